// ChamferDistance_15427522527319
// MI455X (gfx1250) — compile-verified
//
#include <hip/hip_runtime.h>

typedef __attribute__((ext_vector_type(2))) float v2f;
typedef __attribute__((ext_vector_type(8))) float v8f;

#define WAVES_PER_BLOCK 8

// For each point in A[b, 0..Na), compute min squared distance to the set
// B[b, 0..Nb). One wave handles a 16-row tile of A; cross terms via
// V_WMMA_F32_16X16X4_F32 (K=3 zero-padded to 4).
__global__ __launch_bounds__(256) void nn_min_wmma(
    const float* __restrict__ A, const float* __restrict__ Bp,
    float* __restrict__ out, int Na, int Nb)
{
    const int lane = threadIdx.x & 31;
    const int wave = threadIdx.x >> 5;
    const int tilesPerBatch = Na >> 4;
    const int tile = blockIdx.x * WAVES_PER_BLOCK + wave;
    if (tile >= tilesPerBatch) return;   // wave-uniform: EXEC stays all-1s

    const int b = blockIdx.y;
    const float* __restrict__ Abase = A  + (size_t)b * Na * 3;
    const float* __restrict__ Bbase = Bp + (size_t)b * Nb * 3;
    float* __restrict__ obase = out + (size_t)b * Na;

    const int half = lane >> 4;   // 0 -> K=0,1 ; 1 -> K=2,3(pad)
    const int l16  = lane & 15;

    // ---- A fragment: 16x4 f32. Rows M = tile*16 + l16. ----
    const int arow = tile * 16 + l16;
    v2f afrag;
    if (half == 0) {
        afrag.x = Abase[arow * 3 + 0];
        afrag.y = Abase[arow * 3 + 1];
    } else {
        afrag.x = Abase[arow * 3 + 2];
        afrag.y = 0.0f;
    }

    // Running min of (y2_j - 2*cross) for the 8 C-rows this lane holds.
    float minv[8];
    #pragma unroll
    for (int r = 0; r < 8; ++r) minv[r] = 3.402823466e38f;

    const int nbTiles = Nb >> 4;
    #pragma unroll 2
    for (int t = 0; t < nbTiles; ++t) {
        // Column j for this lane (same j for both halves, matching C layout N=l16).
        const int j = t * 16 + l16;
        const float p0 = Bbase[j * 3 + 0];
        const float p1 = Bbase[j * 3 + 1];
        const float p2 = Bbase[j * 3 + 2];
        const float y2 = p0 * p0 + p1 * p1 + p2 * p2;

        // ---- B fragment: 4x16 f32. Cols N = l16, K split across halves. ----
        v2f bfrag;
        if (half == 0) { bfrag.x = p0; bfrag.y = p1; }
        else           { bfrag.x = p2; bfrag.y = 0.0f; }

        v8f c = {};
        // 8 args: (neg_a, A, neg_b, B, c_mod, C, reuse_a, reuse_b)
        c = __builtin_amdgcn_wmma_f32_16x16x4_f32(
                false, afrag, false, bfrag, (short)0, c, false, false);

        #pragma unroll
        for (int r = 0; r < 8; ++r) {
            // d_partial = y2_j - 2*cross(i,j)   (x2_i added once at the end)
            const float d = __builtin_fmaf(-2.0f, c[r], y2);
            minv[r] = fminf(minv[r], d);
        }
    }

    // ---- Min across the 16 lanes of each half (halves hold different rows,
    //      so masks stay <16 to avoid crossing halves). ----
    #pragma unroll
    for (int r = 0; r < 8; ++r) {
        float m = minv[r];
        m = fminf(m, __shfl_xor(m, 1, 32));
        m = fminf(m, __shfl_xor(m, 2, 32));
        m = fminf(m, __shfl_xor(m, 4, 32));
        m = fminf(m, __shfl_xor(m, 8, 32));
        minv[r] = m;
    }

    // Lane 0 holds rows 0..7, lane 16 holds rows 8..15.
    if (l16 == 0) {
        #pragma unroll
        for (int r = 0; r < 8; ++r) {
            const int row = tile * 16 + half * 8 + r;
            const float q0 = Abase[row * 3 + 0];
            const float q1 = Abase[row * 3 + 1];
            const float q2 = Abase[row * 3 + 2];
            const float x2 = q0 * q0 + q1 * q1 + q2 * q2;
            obase[row] = x2 + minv[r];
        }
    }
}

extern "C" void kernel_launch(void* const* d_in, const int* in_sizes, int n_in,
                              void* d_out, int out_size, void* d_ws, size_t ws_size,
                              hipStream_t stream) {
    (void)n_in; (void)d_ws; (void)ws_size; (void)out_size;
    const float* xyz1 = (const float*)d_in[0];
    const float* xyz2 = (const float*)d_in[1];
    float* out = (float*)d_out;

    const int B = 8;
    const int N = in_sizes[0] / (B * 3);   // 8192
    const int M = in_sizes[1] / (B * 3);   // 8192

    dim3 block(32 * WAVES_PER_BLOCK);

    // dist1: for each xyz1 point, min over xyz2
    dim3 grid1(((N / 16) + WAVES_PER_BLOCK - 1) / WAVES_PER_BLOCK, B);
    nn_min_wmma<<<grid1, block, 0, stream>>>(xyz1, xyz2, out, N, M);

    // dist2: same kernel, swapped operands
    dim3 grid2(((M / 16) + WAVES_PER_BLOCK - 1) / WAVES_PER_BLOCK, B);
    nn_min_wmma<<<grid2, block, 0, stream>>>(xyz2, xyz1, out + (size_t)B * N, M, N);
}